// WaveGeneratorOscillator_10548439679011
// MI455X (gfx1250) — compile-verified
//
#include <hip/hip_runtime.h>
#include <math.h>

// ---------------- problem constants (from reference) ----------------
#define B_        8
#define T_SAMP    64000
#define H_        100
#define NYQ       8000.0f
#define DPHI      3.92699081698724139e-4f   /* 2*pi/16000 */
#define TWO_PI_F  6.283185307179586476925f

// ---------------- tiling ----------------
#define TILE      1024
#define NT        63          /* ceil(64000/1024): 62 full + one 512 tile */
#define BLOCK     256         /* 8 wave32 per block, 4 samples per thread */

// workspace layout (floats)
#define WS_SUMS   0
#define WS_OFFS   1024

typedef unsigned int  v4u __attribute__((ext_vector_type(4)));
typedef int           v4i __attribute__((ext_vector_type(4)));
typedef int           v8i __attribute__((ext_vector_type(8)));

#if defined(__has_builtin)
#  if __has_builtin(__builtin_amdgcn_tensor_load_to_lds)
#    define HAVE_TDM 1
#  endif
#  if __has_builtin(__builtin_amdgcn_s_wait_tensorcnt)
#    define HAVE_WAIT_TENSOR 1
#  endif
#endif

__device__ __forceinline__ void wait_tensorcnt0() {
#if defined(HAVE_WAIT_TENSOR)
  __builtin_amdgcn_s_wait_tensorcnt(0);
#else
  asm volatile("s_wait_tensorcnt 0" ::: "memory");
#endif
}

#if defined(HAVE_TDM)
// 1-D f32 TDM load: tensor_n valid elements in memory, tile_n elements moved
// into LDS (elements past tensor_n are zero-filled by the TDM -> handles the
// ragged last tile with no branches).  Descriptor per CDNA5 ISA ch.8.
__device__ __forceinline__ void tdm_load_1d_f32(unsigned ldsOff, const float* gptr,
                                                unsigned tensor_n, unsigned tile_n) {
  unsigned long long ga = (unsigned long long)(const void*)gptr;
  v4u g0;
  g0[0] = 1u;                                            // count=1, user mode
  g0[1] = ldsOff;                                        // lds_addr (bytes)
  g0[2] = (unsigned)(ga & 0xFFFFFFFFull);                // global_addr[31:0]
  g0[3] = (unsigned)((ga >> 32) & 0x1FFFFFFull)          // global_addr[56:32]
        | (2u << 30);                                    // type = 2 ("image")
  v8i g1;
  g1[0] = (int)(2u << 16);                               // data_size=2 (4B); wg_mask=0
  g1[1] = (int)((tensor_n & 0xFFFFu) << 16);             // tensor_dim0[15:0]
  g1[2] = (int)(((tensor_n >> 16) & 0xFFFFu)             // tensor_dim0[31:16]
        | (1u << 16));                                   // tensor_dim1 = 1
  g1[3] = (int)((tile_n & 0xFFFFu) << 16);               // tile_dim0
  g1[4] = 0;                                             // tile_dim1/2 unused
  g1[5] = (int)tensor_n;                                 // tensor_dim0_stride lo
  g1[6] = 0;
  g1[7] = 0;
  v4i z4 = (v4i)0;
#if __clang_major__ >= 23
  v8i z8 = (v8i)0;
  __builtin_amdgcn_tensor_load_to_lds(g0, g1, z4, z4, z8, 0);
#else
  __builtin_amdgcn_tensor_load_to_lds(g0, g1, z4, z4, 0);
#endif
}
#endif

// ---------------- kernel 1: per-tile sums of dphi ----------------
__global__ void osc_tile_sums(const float* __restrict__ f0, float* __restrict__ ws) {
  const int tile = blockIdx.x, b = blockIdx.y;
  const int tid  = threadIdx.x;
  const int base = tile * TILE + tid * 4;
  float s = 0.0f;
  if (base + 3 < T_SAMP) {
    float4 v = *reinterpret_cast<const float4*>(f0 + (size_t)b * T_SAMP + base);
    s = (v.x + v.y) + (v.z + v.w);
  }
  s *= DPHI;
  for (int d = 16; d > 0; d >>= 1) s += __shfl_down(s, d, 32);
  __shared__ float wsum[BLOCK / 32];
  if ((tid & 31) == 0) wsum[tid >> 5] = s;
  __syncthreads();
  if (tid == 0) {
    float t = 0.0f;
#pragma unroll
    for (int w = 0; w < BLOCK / 32; ++w) t += wsum[w];
    ws[WS_SUMS + b * NT + tile] = t;
  }
}

// ---------------- kernel 2: exclusive scan of tile sums (+init phase) ------
__global__ void osc_scan_tiles(const float* __restrict__ initial_phase,
                               float* __restrict__ ws) {
  const int b = threadIdx.x;
  if (b < B_) {
    float run = initial_phase[b];
    for (int t = 0; t < NT; ++t) {
      ws[WS_OFFS + b * NT + t] = run;
      run += ws[WS_SUMS + b * NT + t];
    }
  }
}

// ---------------- kernel 3: phase scan + Chebyshev harmonic sum ------------
__global__ void __launch_bounds__(BLOCK)
osc_main(const float* __restrict__ f0, const float* __restrict__ amps,
         const float* __restrict__ ratio, const float* __restrict__ ws,
         float* __restrict__ out) {
  const int tile = blockIdx.x, b = blockIdx.y;
  const int tid  = threadIdx.x;
  const int lane = tid & 31, wave = tid >> 5;

  __shared__ __align__(16) float s_f0[TILE];
  __shared__ float s_amp[H_ + 28];
  __shared__ float s_wsum[BLOCK / 32];

  const int tileStart = tile * TILE;
  const int validN    = min(TILE, T_SAMP - tileStart);

#if defined(HAVE_TDM)
  if (tid < 32) {                            // one wave issues both TDM ops
    tdm_load_1d_f32((unsigned)(unsigned long long)(void*)s_f0,
                    f0 + (size_t)b * T_SAMP + tileStart,
                    (unsigned)validN, (unsigned)TILE);
    tdm_load_1d_f32((unsigned)(unsigned long long)(void*)s_amp,
                    amps, (unsigned)H_, (unsigned)H_);
    wait_tensorcnt0();
  }
#else
  for (int i = tid; i < H_; i += BLOCK) s_amp[i] = amps[i];
  {
    float4 v = make_float4(0.f, 0.f, 0.f, 0.f);
    const int base = tileStart + tid * 4;
    if (base + 3 < T_SAMP)
      v = *reinterpret_cast<const float4*>(f0 + (size_t)b * T_SAMP + base);
    reinterpret_cast<float4*>(s_f0)[tid] = v;
  }
#endif
  __syncthreads();

  // ---- phase: in-tile inclusive scan (4 serial per thread + wave32 scan) ----
  const float4 fv = reinterpret_cast<const float4*>(s_f0)[tid];
  const float d0 = fv.x * DPHI, d1 = fv.y * DPHI, d2 = fv.z * DPHI, d3 = fv.w * DPHI;
  const float p0 = d0, p1 = p0 + d1, p2 = p1 + d2, p3 = p2 + d3;

  float inc = p3;
#pragma unroll
  for (int d = 1; d < 32; d <<= 1) {
    float y = __shfl_up(inc, d, 32);
    if (lane >= d) inc += y;
  }
  const float exclT = inc - p3;
  if (lane == 31) s_wsum[wave] = inc;
  __syncthreads();
  float woff = 0.0f;
  for (int w = 0; w < wave; ++w) woff += s_wsum[w];

  const float tileOff = ws[WS_OFFS + b * NT + tile];   // includes initial_phase
  const float baseph  = tileOff + woff + exclT;
  const float ph0 = baseph + p0, ph1 = baseph + p1,
              ph2 = baseph + p2, ph3 = baseph + p3;

  // ---- harmonic sum via Chebyshev recurrence, 4 samples interleaved ----
  const float ratio_s = ratio[0];
  int km0 = 0, km1 = 0, km2 = 0, km3 = 0;
  {
    const float f[4] = {fv.x, fv.y, fv.z, fv.w};
    int* km[4] = {&km0, &km1, &km2, &km3};
#pragma unroll
    for (int j = 0; j < 4; ++j) {
      if (f[j] > 0.0f) {
        int k = (int)(NYQ / f[j]);
        if ((float)k * f[j] >= NYQ) --k;
        else if ((float)(k + 1) * f[j] < NYQ) ++k;
        k = k < 0 ? 0 : (k > H_ ? H_ : k);
        *km[j] = k;
      }
    }
  }
  const int kM = max(max(km0, km1), max(km2, km3));

  float s0, c0, s1, c1, s2, c2, s3, c3;
  sincosf(ph0, &s0, &c0);  sincosf(ph1, &s1, &c1);
  sincosf(ph2, &s2, &c2);  sincosf(ph3, &s3, &c3);
  const float tc0 = 2.f * c0, tc1 = 2.f * c1, tc2 = 2.f * c2, tc3 = 2.f * c3;
  float sp0 = 0.f, sp1 = 0.f, sp2 = 0.f, sp3 = 0.f;      // sin(0*θ)
  float cu0 = s0,  cu1 = s1,  cu2 = s2,  cu3 = s3;       // sin(1*θ)
  float a0c = 0.f, a1c = 0.f, a2c = 0.f, a3c = 0.f;

  for (int k = 1; k <= kM; ++k) {
    const float a = s_amp[k - 1] * ratio_s;
    const float m0 = (k <= km0) ? a : 0.f;
    const float m1 = (k <= km1) ? a : 0.f;
    const float m2 = (k <= km2) ? a : 0.f;
    const float m3 = (k <= km3) ? a : 0.f;
    a0c = fmaf(m0, cu0, a0c);  a1c = fmaf(m1, cu1, a1c);
    a2c = fmaf(m2, cu2, a2c);  a3c = fmaf(m3, cu3, a3c);
    const float n0 = fmaf(tc0, cu0, -sp0);
    const float n1 = fmaf(tc1, cu1, -sp1);
    const float n2 = fmaf(tc2, cu2, -sp2);
    const float n3 = fmaf(tc3, cu3, -sp3);
    sp0 = cu0; cu0 = n0;  sp1 = cu1; cu1 = n1;
    sp2 = cu2; cu2 = n2;  sp3 = cu3; cu3 = n3;
  }

  // ---- store signal ----
  const int gbase = tileStart + tid * 4;
  if (gbase + 3 < T_SAMP) {
    *reinterpret_cast<float4*>(out + (size_t)b * T_SAMP + gbase) =
        make_float4(a0c, a1c, a2c, a3c);
  }
  // ---- final phase: phase at t = T-1, mod 2π ----
  if (gbase + 3 == T_SAMP - 1) {
    out[(size_t)B_ * T_SAMP + b] = fmodf(ph3, TWO_PI_F);
  }
}

// ---------------- launcher ----------------
extern "C" void kernel_launch(void* const* d_in, const int* in_sizes, int n_in,
                              void* d_out, int out_size, void* d_ws, size_t ws_size,
                              hipStream_t stream) {
  const float* f0            = (const float*)d_in[0];   // (B,T,1)
  const float* initial_phase = (const float*)d_in[1];   // (B,1,1)
  const float* amplitudes    = (const float*)d_in[2];   // (H,)
  const float* ratio         = (const float*)d_in[3];   // (1,)
  float*       out           = (float*)d_out;           // signal (B,T) ++ final_phase (B)
  float*       ws            = (float*)d_ws;            // >= (1024+504)*4 bytes

  (void)in_sizes; (void)n_in; (void)out_size; (void)ws_size;

  dim3 grid(NT, B_);
  osc_tile_sums<<<grid, BLOCK, 0, stream>>>(f0, ws);
  osc_scan_tiles<<<1, 32, 0, stream>>>(initial_phase, ws);
  osc_main<<<grid, BLOCK, 0, stream>>>(f0, amplitudes, ratio, ws, out);
}